// SIGN_4320737100473
// MI455X (gfx1250) — compile-verified
//
#include <hip/hip_runtime.h>

#define NN 100000
#define NE 1200000
#define NF 64
#define NH 32
#define NC 40
#define NT (NN / 16)   // 6250 tiles of 16 nodes

typedef float v2f __attribute__((ext_vector_type(2)));
typedef float v8f __attribute__((ext_vector_type(8)));

static __device__ __forceinline__ v8f wmma_f32(v2f a, v2f b, v8f c) {
  // D = A(16x4 f32) x B(4x16 f32) + C(16x16 f32)
  return __builtin_amdgcn_wmma_f32_16x16x4_f32(false, a, false, b, (short)0, c,
                                               false, false);
}

// ---------------- degree / normalization ----------------
__global__ void k_init_deg(float* __restrict__ deg) {
  unsigned i = blockIdx.x * 256u + threadIdx.x;
  if (i < NN) deg[i] = 1.0f;  // self loop
}

__global__ void k_accum_deg(const int* __restrict__ dst, float* __restrict__ deg) {
  unsigned i = blockIdx.x * 256u + threadIdx.x;
  if (i < NE) atomicAdd(&deg[dst[i]], 1.0f);
}

__global__ void k_finalize_dinv(float* __restrict__ deg) {
  unsigned i = blockIdx.x * 256u + threadIdx.x;
  if (i < NN) deg[i] = rsqrtf(deg[i]);   // deg >= 1 always
}

// out[i] = dinv[i]^2 * h[i]   (self-loop contribution, also zero-initializes out)
__global__ void k_selfloop(const float* __restrict__ h, const float* __restrict__ dinv,
                           float* __restrict__ o) {
  unsigned i = blockIdx.x * 256u + threadIdx.x;
  if (i < (unsigned)NN * NF) {
    float d = dinv[i >> 6];
    o[i] = d * d * h[i];
  }
}

// out[dst,f] += dinv[src]*dinv[dst]*h[src,f]  -- 64 lanes per edge (f = tid&63)
__global__ void k_prop(const int* __restrict__ src, const int* __restrict__ dst,
                       const float* __restrict__ dinv, const float* __restrict__ h,
                       float* __restrict__ o) {
  unsigned i = blockIdx.x * 256u + threadIdx.x;
  unsigned e = i >> 6, f = i & 63u;
  if (e < NE) {
    unsigned s = (unsigned)src[e], d = (unsigned)dst[e];
    float w = dinv[s] * dinv[d];
    atomicAdd(o + d * (unsigned)NF + f, w * h[s * (unsigned)NF + f]);
  }
}

// ---------------- fused hidden GEMMs + concat + classifier + log_softmax ----
__global__ __launch_bounds__(256) void k_classify(
    const float* __restrict__ h1, const float* __restrict__ h2,
    const float* __restrict__ h3,
    const float* __restrict__ W1, const float* __restrict__ b1,
    const float* __restrict__ W2, const float* __restrict__ b2,
    const float* __restrict__ W3, const float* __restrict__ b3,
    const float* __restrict__ Wl, const float* __restrict__ bl,
    float* __restrict__ out) {
  // per-wave staging tile: cat features [16][96] (padded), later reused for logits
  __shared__ float smem[8][16][97];

  const int wave = threadIdx.x >> 5;
  const int lane = threadIdx.x & 31;
  int tile = blockIdx.x * 8 + wave;
  if (tile >= NT) tile = NT - 1;  // duplicate last tile (identical writes, benign)
  const int row = lane & 15;      // A/B/C matrix row/col index for this lane
  const int half = lane >> 4;     // K-pair select within k-block of 4
  const int node0 = tile * 16;

  const float* Hs[3] = {h1, h2, h3};
  const float* Ws[3] = {W1, W2, W3};
  const float* bs[3] = {b1, b2, b3};

#pragma unroll
  for (int j = 0; j < 3; ++j) {
    const float* H = Hs[j];
    const float* W = Ws[j];
    v8f acc0 = {}, acc1 = {};
#pragma unroll
    for (int kk = 0; kk < NF / 4; ++kk) {
      const int k = kk * 4 + half * 2;
      // A 16x4 f32: lane holds row `row`, K = {k, k+1}
      v2f a = *(const v2f*)(H + (unsigned)(node0 + row) * NF + k);
      // B 4x16 f32: lane holds col `row` of rows K = {k, k+1}; two 16-col tiles
      v2f bn0, bn1;
      bn0.x = W[k * NH + row];
      bn0.y = W[(k + 1) * NH + row];
      bn1.x = W[k * NH + 16 + row];
      bn1.y = W[(k + 1) * NH + 16 + row];
      acc0 = wmma_f32(a, bn0, acc0);
      acc1 = wmma_f32(a, bn1, acc1);
    }
    // C/D layout: VGPR v, lanes 0-15 -> M=v, lanes 16-31 -> M=v+8; N = lane&15
    const float bias0 = bs[j][row];
    const float bias1 = bs[j][16 + row];
#pragma unroll
    for (int v = 0; v < 8; ++v) {
      const int m = v + half * 8;
      smem[wave][m][j * NH + row]      = fmaxf(acc0[v] + bias0, 0.0f);
      smem[wave][m][j * NH + 16 + row] = fmaxf(acc1[v] + bias1, 0.0f);
    }
  }
  __syncthreads();

  // cat[16x96] @ lin_W[96x40] via WMMA; 3 n-tiles of 16 (cols 40..47 zero-padded)
  v8f acc[3] = {v8f{}, v8f{}, v8f{}};
#pragma unroll
  for (int kk = 0; kk < (3 * NH) / 4; ++kk) {
    const int k = kk * 4 + half * 2;
    v2f a;
    a.x = smem[wave][row][k];
    a.y = smem[wave][row][k + 1];
#pragma unroll
    for (int t = 0; t < 3; ++t) {
      const int n = t * 16 + row;
      v2f bb;
      bb.x = (n < NC) ? Wl[k * NC + n] : 0.0f;
      bb.y = (n < NC) ? Wl[(k + 1) * NC + n] : 0.0f;
      acc[t] = wmma_f32(a, bb, acc[t]);
    }
  }
  __syncthreads();

  // logits (+bias) back into LDS (reuse cat region; cat fully consumed above)
#pragma unroll
  for (int t = 0; t < 3; ++t) {
    const int n = t * 16 + row;
    const float bias = (n < NC) ? bl[n] : 0.0f;
#pragma unroll
    for (int v = 0; v < 8; ++v) {
      const int m = v + half * 8;
      smem[wave][m][n] = acc[t][v] + bias;
    }
  }
  __syncthreads();

  // log_softmax: lanes 0..15 each own one row of 40 logits
  if (lane < 16) {
    const int node = node0 + lane;
    float mx = -3.4e38f;
    for (int c = 0; c < NC; ++c) mx = fmaxf(mx, smem[wave][lane][c]);
    float s = 0.0f;
    for (int c = 0; c < NC; ++c) s += expf(smem[wave][lane][c] - mx);
    const float lse = mx + logf(s);
    for (int c = 0; c < NC; ++c) out[(unsigned)node * NC + c] = smem[wave][lane][c] - lse;
  }
}

extern "C" void kernel_launch(void* const* d_in, const int* in_sizes, int n_in,
                              void* d_out, int out_size, void* d_ws, size_t ws_size,
                              hipStream_t stream) {
  const float* x  = (const float*)d_in[0];
  const int*   ei = (const int*)d_in[1];   // [2, NE]: src row then dst row
  const float* W1 = (const float*)d_in[2];
  const float* b1 = (const float*)d_in[3];
  const float* W2 = (const float*)d_in[4];
  const float* b2 = (const float*)d_in[5];
  const float* W3 = (const float*)d_in[6];
  const float* b3 = (const float*)d_in[7];
  const float* Wl = (const float*)d_in[8];
  const float* bl = (const float*)d_in[9];
  float* out = (float*)d_out;

  const int* src = ei;
  const int* dst = ei + NE;

  // workspace: dinv [NN] (padded to 102400), h2 [NN*NF], h3 [NN*NF]  ~51.6 MB
  float* ws   = (float*)d_ws;
  float* dinv = ws;
  float* h2   = ws + 102400;
  float* h3   = h2 + (size_t)NN * NF;

  k_init_deg<<<(NN + 255) / 256, 256, 0, stream>>>(dinv);
  k_accum_deg<<<(NE + 255) / 256, 256, 0, stream>>>(dst, dinv);
  k_finalize_dinv<<<(NN + 255) / 256, 256, 0, stream>>>(dinv);

  // hop 1: h2 = prop(x)
  k_selfloop<<<(NN * NF) / 256, 256, 0, stream>>>(x, dinv, h2);
  k_prop<<<(NE * 64) / 256, 256, 0, stream>>>(src, dst, dinv, x, h2);
  // hop 2: h3 = prop(h2)
  k_selfloop<<<(NN * NF) / 256, 256, 0, stream>>>(h2, dinv, h3);
  k_prop<<<(NE * 64) / 256, 256, 0, stream>>>(src, dst, dinv, h2, h3);

  // fused dense head
  k_classify<<<(NT + 7) / 8, 256, 0, stream>>>(x, h2, h3, W1, b1, W2, b2, W3, b3,
                                               Wl, bl, out);
}